// VectorQuantizer_30296699306285
// MI455X (gfx1250) — compile-verified
//
#include <hip/hip_runtime.h>

typedef __attribute__((ext_vector_type(2))) float v2f;
typedef __attribute__((ext_vector_type(8))) float v8f;

#define D_DIM 64
#define KSTEPS 16            // D / 4 (wmma f32 k-step is 4)
#define ROWS_PER_WAVE 16
#define ROWS_PER_BLOCK 128   // 8 waves * 16 rows
#define MAX_K 1024

// order-preserving float <-> uint encoding for atomicMin on distances
__device__ __forceinline__ unsigned fenc(float f) {
    unsigned u = __float_as_uint(f);
    return (u & 0x80000000u) ? ~u : (u | 0x80000000u);
}
__device__ __forceinline__ float fdec(unsigned k) {
    unsigned u = (k & 0x80000000u) ? (k & 0x7fffffffu) : ~k;
    return __uint_as_float(u);
}

__global__ void vq_init_kernel(unsigned* __restrict__ gcolmin,
                               float* __restrict__ sumsq, int K) {
    int i = blockIdx.x * 256 + threadIdx.x;
    if (i < K) gcolmin[i] = 0xFFFFFFFFu;
    if (i == 0) *sumsq = 0.0f;
}

// Precompute ||c_k||^2 so the hot GEMM loop has no load-dependent FMA chain.
__global__ void vq_cnorm_kernel(const float* __restrict__ cb,
                                float* __restrict__ gcn, int K) {
    int k = blockIdx.x * 256 + threadIdx.x;
    if (k >= K) return;
    const float4* r = (const float4*)(cb + (size_t)k * D_DIM);
    float s = 0.0f;
#pragma unroll
    for (int j = 0; j < D_DIM / 4; ++j) {
        float4 v = r[j];
        s += v.x * v.x + v.y * v.y + v.z * v.z + v.w * v.w;
    }
    gcn[k] = s;
}

// ---- tile helpers for the main kernel ----
__device__ __forceinline__ void load_tile(const float* __restrict__ cb,
                                          const float* __restrict__ gcn,
                                          int colbase, int mlo, int hi,
                                          v2f b[KSTEPS], float& cn) {
    const float* crow = cb + (size_t)(colbase + mlo) * D_DIM;
#pragma unroll
    for (int kk = 0; kk < KSTEPS; ++kk)
        b[kk] = *(const v2f*)(crow + 4 * kk + 2 * hi);
    cn = gcn[colbase + mlo];
}

__device__ __forceinline__ void process_tile(const v2f a[KSTEPS],
                                             const v2f b[KSTEPS], float cn,
                                             const float xn[8], int colbase,
                                             int mlo, float best_d[8],
                                             int best_i[8],
                                             unsigned* __restrict__ s_colmin) {
    // two independent accumulation chains for XDL ILP
    v8f acc0 = {}, acc1 = {};
#pragma unroll
    for (int kk = 0; kk < KSTEPS / 2; ++kk) {
        acc0 = __builtin_amdgcn_wmma_f32_16x16x4_f32(
            false, a[kk], false, b[kk], (short)0, acc0, false, false);
        acc1 = __builtin_amdgcn_wmma_f32_16x16x4_f32(
            false, a[kk + 8], false, b[kk + 8], (short)0, acc1, false, false);
    }
    const int col = colbase + mlo;
    float cmin = 3.4e38f;
#pragma unroll
    for (int v = 0; v < 8; ++v) {
        float s = acc0[v] + acc1[v];
        float dist = xn[v] - 2.0f * s + cn;
        if (dist < best_d[v]) { best_d[v] = dist; best_i[v] = col; }
        cmin = fminf(cmin, dist);
    }
    // column-min over this wave's 16 rows (combine the two half-waves)
    cmin = fminf(cmin, __shfl_xor(cmin, 16, 32));
    atomicMin(&s_colmin[col], fenc(cmin));
}

// Main fused kernel: FP32 WMMA GEMM (x @ cbT) + per-row argmin + per-code min.
// grid = N/128 blocks of 256 threads (8 waves, 16 rows each).
__global__ void __launch_bounds__(256)
vq_argmin_kernel(const float* __restrict__ x, const float* __restrict__ cb,
                 const float* __restrict__ gcn, int* __restrict__ tokens,
                 unsigned* __restrict__ gcolmin, int K)
{
    __shared__ unsigned s_colmin[MAX_K];
    const int tid = threadIdx.x;
    for (int i = tid; i < K; i += 256) s_colmin[i] = 0xFFFFFFFFu;
    __syncthreads();

    const int wave = tid >> 5;
    const int lane = tid & 31;
    const int mlo  = lane & 15;   // M (for A) / N (for B,C) index within tile
    const int hi   = lane >> 4;   // half-wave select
    const int rowbase = blockIdx.x * ROWS_PER_BLOCK + wave * ROWS_PER_WAVE;

    // ---- load A fragments for this wave's 16 rows (held in registers) ----
    // ISA layout A 16x4 f32: lane L -> M = L%16, VGPR pair = K = 4kk + 2*hi + {0,1}
    v2f a[KSTEPS];
    float xn_half = 0.0f;
    const float* xrow = x + (size_t)(rowbase + mlo) * D_DIM;
#pragma unroll
    for (int kk = 0; kk < KSTEPS; ++kk) {
        a[kk] = *(const v2f*)(xrow + 4 * kk + 2 * hi);
        xn_half += a[kk].x * a[kk].x + a[kk].y * a[kk].y;
    }
    // full ||x_row||^2: the two half-waves each hold half the row's elements
    float xnorm = xn_half + __shfl_xor(xn_half, 16, 32);
    // acc element (v, lane) is row m = v + 8*hi -> fetch that row's norm
    float xn[8];
#pragma unroll
    for (int v = 0; v < 8; ++v)
        xn[v] = __shfl(xnorm, v + 8 * hi, 32);

    float best_d[8];
    int   best_i[8];
#pragma unroll
    for (int v = 0; v < 8; ++v) { best_d[v] = 3.4e38f; best_i[v] = 0; }

    // ---- sweep all K codes, 16-wide tiles, double-buffered B (ping/pong) ----
    const int ntiles = K >> 4;   // 64, even
    v2f bA[KSTEPS], bB[KSTEPS];
    float cnA, cnB;
    load_tile(cb, gcn, 0, mlo, hi, bA, cnA);
    for (int c0 = 0; c0 < ntiles; c0 += 2) {
        load_tile(cb, gcn, (c0 + 1) << 4, mlo, hi, bB, cnB);
        process_tile(a, bA, cnA, xn, c0 << 4, mlo, best_d, best_i, s_colmin);
        if (c0 + 2 < ntiles)
            load_tile(cb, gcn, (c0 + 2) << 4, mlo, hi, bA, cnA);
        process_tile(a, bB, cnB, xn, (c0 + 1) << 4, mlo, best_d, best_i, s_colmin);
    }

    // ---- per-row argmin: butterfly across the 16 lanes of each half-wave ----
#pragma unroll
    for (int off = 1; off < 16; off <<= 1) {
#pragma unroll
        for (int v = 0; v < 8; ++v) {
            float od = __shfl_xor(best_d[v], off, 32);
            int   oi = __shfl_xor(best_i[v], off, 32);
            if (od < best_d[v] || (od == best_d[v] && oi < best_i[v])) {
                best_d[v] = od; best_i[v] = oi;
            }
        }
    }
    if (mlo == 0) {
#pragma unroll
        for (int v = 0; v < 8; ++v)
            tokens[rowbase + 8 * hi + v] = best_i[v];
    }

    __syncthreads();
    for (int i = tid; i < K; i += 256)
        atomicMin(&gcolmin[i], s_colmin[i]);
}

// Gather emb = codebook[token], write output, accumulate sum((emb-x)^2).
__global__ void __launch_bounds__(256)
vq_gather_kernel(const float* __restrict__ x, const float* __restrict__ cb,
                 const int* __restrict__ tokens, float* __restrict__ out,
                 float* __restrict__ sumsq, int n_vec4)
{
    int e = blockIdx.x * 256 + threadIdx.x;
    float sq = 0.0f;
    if (e < n_vec4) {
        int row = e >> 4, j = e & 15;  // 16 float4 per row (D=64)
        int tok = tokens[row];
        const float4* cb4 = (const float4*)cb;
        const float4* x4  = (const float4*)x;
        float4 c  = cb4[tok * 16 + j];
        float4 xi = x4[e];
        ((float4*)out)[e] = c;
        float dx = c.x - xi.x, dy = c.y - xi.y, dz = c.z - xi.z, dw = c.w - xi.w;
        sq = dx * dx + dy * dy + dz * dz + dw * dw;
    }
#pragma unroll
    for (int o = 16; o > 0; o >>= 1) sq += __shfl_xor(sq, o, 32);
    if ((threadIdx.x & 31) == 0) atomicAdd(sumsq, sq);
}

// Final scalar loss = 1.25*mean((emb-x)^2) + 0.1*mean_k(min_n dist)
__global__ void __launch_bounds__(1024)
vq_loss_kernel(const unsigned* __restrict__ gcolmin,
               const float* __restrict__ sumsq,
               float* __restrict__ loss_out, int K, float inv_ND)
{
    __shared__ float s[32];
    int tid = threadIdx.x;
    float v = (tid < K) ? fdec(gcolmin[tid]) : 0.0f;
#pragma unroll
    for (int o = 16; o > 0; o >>= 1) v += __shfl_xor(v, o, 32);
    if ((tid & 31) == 0) s[tid >> 5] = v;
    __syncthreads();
    if (tid == 0) {
        float tot = 0.0f;
        for (int w = 0; w < 32; ++w) tot += s[w];
        *loss_out = 1.25f * sumsq[0] * inv_ND + 0.1f * tot / (float)K;
    }
}

extern "C" void kernel_launch(void* const* d_in, const int* in_sizes, int n_in,
                              void* d_out, int out_size, void* d_ws, size_t ws_size,
                              hipStream_t stream) {
    const float* x  = (const float*)d_in[0];   // [B,H,W,D] flat, N*D floats
    const float* cb = (const float*)d_in[1];   // [K,D] floats
    const int ND = in_sizes[0];                // 2097152
    const int N  = ND / D_DIM;                 // 32768
    const int K  = in_sizes[1] / D_DIM;        // 1024
    float* out = (float*)d_out;                // N*D outputs + 1 loss scalar

    int*      tokens  = (int*)d_ws;
    unsigned* gcolmin = (unsigned*)((char*)d_ws + (size_t)N * sizeof(int));
    float*    sumsq   = (float*)((char*)gcolmin + (size_t)K * sizeof(unsigned));
    float*    gcn     = (float*)((char*)sumsq + sizeof(float));

    vq_init_kernel<<<(K + 255) / 256, 256, 0, stream>>>(gcolmin, sumsq, K);
    vq_cnorm_kernel<<<(K + 255) / 256, 256, 0, stream>>>(cb, gcn, K);
    vq_argmin_kernel<<<N / ROWS_PER_BLOCK, 256, 0, stream>>>(x, cb, gcn, tokens,
                                                             gcolmin, K);
    const int nvec4 = ND / 4;
    vq_gather_kernel<<<(nvec4 + 255) / 256, 256, 0, stream>>>(x, cb, tokens, out,
                                                              sumsq, nvec4);
    vq_loss_kernel<<<1, 1024, 0, stream>>>(gcolmin, sumsq, out + ND, K,
                                           1.0f / (float)ND);
}